// PairwiseMaxMarginHingeLossFunc2_21775484190867
// MI455X (gfx1250) — compile-verified
//
#include <hip/hip_runtime.h>

typedef __attribute__((ext_vector_type(2))) float v2f;
typedef __attribute__((ext_vector_type(8))) float v8f;

#define NN 256
#define BB 8
#define ROWS (NN * BB) /* 2048 rows: one per (batch, i) */

// One wave (32 lanes) handles one (b,i) row: computes
//   sum_{j,k} relu(margin - (u_j-u_k)(v_j-v_k))
// via 256 V_WMMA_F32_16X16X4_F32 tiles using the rank-4 identity
//   margin - (u_j-u_k)(v_j-v_k) = [u_j,v_j,c_j,1] . [v_k,u_k,-1,margin-c_k]
// plus the argmax(u)==argmax(v) match flag.
__global__ __launch_bounds__(256) void pm_rows_kernel(
    const float* __restrict__ om, const float* __restrict__ lm,
    const int* __restrict__ marginp,
    float* __restrict__ loss_part, int* __restrict__ match_part)
{
  __shared__ float sU[8][NN];
  __shared__ float sV[8][NN];
  __shared__ float sC[8][NN];

  const int w    = threadIdx.x >> 5;   // wave within block
  const int lane = threadIdx.x & 31;
  const int row  = blockIdx.x * 8 + w; // row = b*N + i
  const int b    = row >> 8;
  const int i    = row & (NN - 1);
  const float margin = (float)(*marginp);

  // out_matrix is [N, N, B] row-major: elem (i, j, b) at i*N*B + j*B + b
  const int base = i * (NN * BB) + b;

  float bestU = -__builtin_inff(); int idxU = 0;
  float bestV = -__builtin_inff(); int idxV = 0;

#pragma unroll
  for (int s = 0; s < 8; ++s) {
    const int j  = s * 32 + lane;     // ascending j within each lane
    const float uv = om[base + j * BB];
    const float vv = lm[base + j * BB];
    sU[w][j] = uv;
    sV[w][j] = vv;
    sC[w][j] = uv * vv;
    if (uv > bestU) { bestU = uv; idxU = j; }  // strict > keeps first occurrence
    if (vv > bestV) { bestV = vv; idxV = j; }
  }
  __syncthreads();

  // Wave argmax reduction; tie-break to smaller index (jnp.argmax semantics).
#pragma unroll
  for (int off = 16; off > 0; off >>= 1) {
    float ou = __shfl_xor(bestU, off, 32);
    int  oiu = __shfl_xor(idxU,  off, 32);
    if (ou > bestU || (ou == bestU && oiu < idxU)) { bestU = ou; idxU = oiu; }
    float ov = __shfl_xor(bestV, off, 32);
    int  oiv = __shfl_xor(idxV,  off, 32);
    if (ov > bestV || (ov == bestV && oiv < idxV)) { bestV = ov; idxV = oiv; }
  }

  // WMMA 16x16x4 lane layout (wave32):
  //   A (16x4): lane L holds row m=L&15; lanes 0-15 -> K=0 (.x), K=1 (.y);
  //             lanes 16-31 -> K=2 (.x), K=3 (.y)
  //   B (4x16): lane L holds col n=L&15; lanes 0-15 -> K=0 (.x), K=1 (.y);
  //             lanes 16-31 -> K=2 (.x), K=3 (.y)
  const int  n16 = lane & 15;
  const bool hi  = (lane >= 16);

  // Pre-build the 16 B fragments (one per 16-wide k tile) in registers.
  v2f Bf[16];
#pragma unroll
  for (int kt = 0; kt < 16; ++kt) {
    const int k = kt * 16 + n16;
    Bf[kt].x = hi ? -1.0f               : sV[w][k]; // K=2 : K=0
    Bf[kt].y = hi ? (margin - sC[w][k]) : sU[w][k]; // K=3 : K=1
  }

  v8f acc = {};
  for (int jt = 0; jt < 16; ++jt) {
    const int j = jt * 16 + n16;
    v2f Af;
    Af.x = hi ? sC[w][j] : sU[w][j];  // K=2 : K=0
    Af.y = hi ? 1.0f     : sV[w][j];  // K=3 : K=1
#pragma unroll
    for (int kt = 0; kt < 16; ++kt) {
      v8f cz = {};
      v8f d = __builtin_amdgcn_wmma_f32_16x16x4_f32(
          /*neg_a=*/false, Af, /*neg_b=*/false, Bf[kt],
          /*c_mod=*/(short)0, cz, /*reuse_a=*/false, /*reuse_b=*/false);
#pragma unroll
      for (int e = 0; e < 8; ++e)
        acc[e] += fmaxf(d[e], 0.0f);   // relu + accumulate epilogue
    }
  }

  float s = acc[0] + acc[1] + acc[2] + acc[3] + acc[4] + acc[5] + acc[6] + acc[7];
#pragma unroll
  for (int off = 16; off > 0; off >>= 1)
    s += __shfl_xor(s, off, 32);

  if (lane == 0) {
    loss_part[row]  = s;
    match_part[row] = (idxU == idxV) ? 1 : 0;
  }
}

// Deterministic fixed-order final reduction (no atomics -> replay-stable).
__global__ __launch_bounds__(256) void pm_reduce_kernel(
    const float* __restrict__ loss_part, const int* __restrict__ match_part,
    float* __restrict__ out)
{
  __shared__ float sl[256];
  __shared__ int   sm[256];
  const int t = threadIdx.x;
  float ls = 0.0f; int ms = 0;
  for (int k = t; k < ROWS; k += 256) { ls += loss_part[k]; ms += match_part[k]; }
  sl[t] = ls; sm[t] = ms;
  __syncthreads();
  for (int off = 128; off > 0; off >>= 1) {
    if (t < off) { sl[t] += sl[t + off]; sm[t] += sm[t + off]; }
    __syncthreads();
  }
  if (t == 0) { out[0] = sl[0]; out[1] = (float)sm[0]; }
}

extern "C" void kernel_launch(void* const* d_in, const int* in_sizes, int n_in,
                              void* d_out, int out_size, void* d_ws, size_t ws_size,
                              hipStream_t stream) {
  const float* om      = (const float*)d_in[0];
  const float* lm      = (const float*)d_in[1];
  const int*   marginp = (const int*)d_in[2];
  float* out = (float*)d_out;

  float* loss_part  = (float*)d_ws;
  int*   match_part = (int*)((char*)d_ws + ROWS * sizeof(float));

  pm_rows_kernel<<<ROWS / 8, 256, 0, stream>>>(om, lm, marginp, loss_part, match_part);
  pm_reduce_kernel<<<1, 256, 0, stream>>>(loss_part, match_part, out);
}